// RCAlignLoss_58832462021082
// MI455X (gfx1250) — compile-verified
//
#include <hip/hip_runtime.h>
#include <hip/hip_bf16.h>
#include <math.h>
#include <stdint.h>

typedef __attribute__((ext_vector_type(2))) float v2f;
typedef __attribute__((ext_vector_type(4))) float v4f;
typedef __attribute__((ext_vector_type(8))) float v8f;

#define B_ 32
#define N_ 576
#define W_ 32
#define D_ 256

#define LSTR 260            // con LDS row stride (dwords): 260 % 64 == 4 -> conflict-free
#define KC   32             // K-chunk
#define KSTR 36             // img chunk row stride (dwords): 36 % 64 residues all distinct
#define NWAVES 4
#define NTILES (N_ / 16)            // 36 n-tiles
#define CHUNKS (D_ / KC)            // 8 K-chunks per tile
#define TPW    (NTILES / NWAVES)    // 9 tiles per wave
#define QTOT   (TPW * CHUNKS)       // 72 chunks per wave

// LDS byte offset of a generic pointer that points into LDS
// (flat LDS aperture address carries the DS offset in its low 32 bits).
__device__ __forceinline__ uint32_t lds_off(const void* p) {
  return (uint32_t)(uintptr_t)p;
}

// CDNA5 async copy: GLOBAL_LOAD_ASYNC_TO_LDS_B128 (per-lane LDS scatter, ASYNCcnt).
__device__ __forceinline__ void async_ld_b128(uint32_t lds_byte, const void* gptr) {
  asm volatile("global_load_async_to_lds_b128 %0, %1, off"
               :
               : "v"(lds_byte), "v"((uint64_t)(uintptr_t)gptr)
               : "memory");
}

// ---------------------------------------------------------------------------
// Kernel 1: L2-normalize rows of img (B*N x D) and con (B*W x D).
// ---------------------------------------------------------------------------
__global__ void k_normalize(const float* __restrict__ img,
                            const float* __restrict__ con,
                            float* __restrict__ img_n,
                            float* __restrict__ con_n) {
  const int wave = threadIdx.x >> 5;
  const int lane = threadIdx.x & 31;
  const int row  = blockIdx.x * (blockDim.x >> 5) + wave;
  const int nImgRows = B_ * N_;
  const int nTot     = nImgRows + B_ * W_;
  if (row >= nTot) return;

  const float* src;
  float* dst;
  if (row < nImgRows) {
    src = img + (size_t)row * D_;
    dst = img_n + (size_t)row * D_;
  } else {
    const int r = row - nImgRows;
    src = con + (size_t)r * D_;
    dst = con_n + (size_t)r * D_;
  }

  v4f a = *(const v4f*)(src + lane * 8);
  v4f b = *(const v4f*)(src + lane * 8 + 4);
  float ss = a.x * a.x + a.y * a.y + a.z * a.z + a.w * a.w +
             b.x * b.x + b.y * b.y + b.z * b.z + b.w * b.w;
#pragma unroll
  for (int off = 16; off >= 1; off >>= 1) ss += __shfl_xor(ss, off, 32);
  const float s = 1.0f / fmaxf(sqrtf(ss), 1e-12f);
  a *= s;
  b *= s;
  *(v4f*)(dst + lane * 8)     = a;
  *(v4f*)(dst + lane * 8 + 4) = b;
}

// ---------------------------------------------------------------------------
// Kernel 2: per (m,c): maxbuf[m][c][w] = max_n img_n[m,n,:].con_n[c,w,:]
// Async-to-LDS staging (double-buffered) feeding f32 WMMA 16x16x4.
// ---------------------------------------------------------------------------
__global__ __launch_bounds__(128)
void k_gemm_max(const float* __restrict__ img_n,
                const float* __restrict__ con_n,
                float* __restrict__ maxbuf) {
  __shared__ __align__(16) float conLds[W_ * LSTR];               // 33280 B
  __shared__ __align__(16) float imgLds[NWAVES * 2 * 16 * KSTR];  // 18432 B
  __shared__ float partial[NWAVES * W_];                          //   512 B

  const int m    = blockIdx.x >> 5;
  const int c    = blockIdx.x & 31;
  const int tid  = threadIdx.x;
  const int wave = tid >> 5;
  const int lane = tid & 31;

  const float* __restrict__ imgM = img_n + (size_t)m * N_ * D_;
  const float* __restrict__ conC = con_n + (size_t)c * W_ * D_;

  // ---- stage con (32x256) into padded LDS via async copies -----------------
  {
    const uint32_t base = lds_off(conLds);
#pragma unroll
    for (int i = 0; i < 16; ++i) {
      const int idx = i * 128 + tid;
      const int row = idx >> 6;  // 64 float4 per row
      const int c4  = idx & 63;
      async_ld_b128(base + (uint32_t)(row * LSTR + c4 * 4) * 4u,
                    conC + row * D_ + c4 * 4);
    }
    asm volatile("s_wait_asynccnt 0" ::: "memory");
  }
  __syncthreads();

  const int halfK = (lane >> 4) * 2;  // K sub-offset: 0 (lanes 0-15) / 2 (16-31)
  const int rsel  = lane & 15;

  // wave-private double-buffered image slab
  const uint32_t slabOff  = lds_off(imgLds) + (uint32_t)(wave * 2 * 16 * KSTR) * 4u;
  const float*   slabPtr  = &imgLds[wave * 2 * 16 * KSTR];

  // issue the 4 async b128 copies of flat-chunk q into buffer (q&1)
  auto issue_chunk = [&](int q) {
    const int n0 = (wave + (q >> 3) * NWAVES) * 16;
    const int kc = (q & (CHUNKS - 1)) * KC;
    const uint32_t bufB = slabOff + (uint32_t)((q & 1) * 16 * KSTR) * 4u;
#pragma unroll
    for (int i = 0; i < 4; ++i) {
      const int idx = i * 32 + lane;
      const int row = idx >> 3;  // 8 float4 per 32-float row chunk
      const int c4  = idx & 7;
      async_ld_b128(bufB + (uint32_t)(row * KSTR + c4 * 4) * 4u,
                    imgM + (size_t)(n0 + row) * D_ + kc + c4 * 4);
    }
  };

  v8f acc0, acc1, vmax0, vmax1;
#pragma unroll
  for (int r = 0; r < 8; ++r) {
    acc0[r] = 0.0f;
    acc1[r] = 0.0f;
    vmax0[r] = -__builtin_inff();
    vmax1[r] = -__builtin_inff();
  }

  // compute flat-chunk q out of buffer (q&1): 8 K-steps x 2 w-tiles of WMMA
  auto compute_chunk = [&](int q) {
    const float* buf = slabPtr + (q & 1) * 16 * KSTR;
    const int kc = (q & (CHUNKS - 1)) * KC;
    v2f af[8], b0[8], b1[8];
#pragma unroll
    for (int s = 0; s < 8; ++s) {
      af[s] = *(const v2f*)(buf + rsel * KSTR + 4 * s + halfK);
      b0[s] = *(const v2f*)(&conLds[rsel * LSTR + kc + 4 * s + halfK]);
      b1[s] = *(const v2f*)(&conLds[(16 + rsel) * LSTR + kc + 4 * s + halfK]);
    }
#pragma unroll
    for (int s = 0; s < 8; ++s) {
      acc0 = __builtin_amdgcn_wmma_f32_16x16x4_f32(
          false, af[s], false, b0[s], (short)0, acc0, false, false);
      acc1 = __builtin_amdgcn_wmma_f32_16x16x4_f32(
          false, af[s], false, b1[s], (short)0, acc1, false, false);
    }
  };

  auto fold_max = [&]() {
#pragma unroll
    for (int r = 0; r < 8; ++r) {
      vmax0[r] = fmaxf(vmax0[r], acc0[r]);
      vmax1[r] = fmaxf(vmax1[r], acc1[r]);
      acc0[r] = 0.0f;
      acc1[r] = 0.0f;
    }
  };

  // ---- software-pipelined main loop: prefetch q+1 while computing q --------
  issue_chunk(0);
  for (int q = 0; q < QTOT - 1; ++q) {
    issue_chunk(q + 1);
    asm volatile("s_wait_asynccnt 4" ::: "memory");  // chunk q resident
    compute_chunk(q);
    if ((q & (CHUNKS - 1)) == CHUNKS - 1) fold_max();  // tile finished
  }
  asm volatile("s_wait_asynccnt 0" ::: "memory");
  compute_chunk(QTOT - 1);
  fold_max();

  // ---- reduce max over M (patches): 8 C/D VGPRs + lane-half split ----------
  float mx0 = vmax0[0], mx1 = vmax1[0];
#pragma unroll
  for (int r = 1; r < 8; ++r) {
    mx0 = fmaxf(mx0, vmax0[r]);
    mx1 = fmaxf(mx1, vmax1[r]);
  }
  mx0 = fmaxf(mx0, __shfl_xor(mx0, 16, 32));
  mx1 = fmaxf(mx1, __shfl_xor(mx1, 16, 32));

  if (lane < 16) {
    partial[wave * W_ + lane]      = mx0;  // w = lane
    partial[wave * W_ + 16 + lane] = mx1;  // w = 16 + lane
  }
  __syncthreads();

  if (tid < W_) {
    float mx = partial[tid];
#pragma unroll
    for (int wv = 1; wv < NWAVES; ++wv) mx = fmaxf(mx, partial[wv * W_ + tid]);
    maxbuf[((size_t)m * B_ + c) * W_ + tid] = mx;
  }
}

// ---------------------------------------------------------------------------
// Kernel 3: ragged masked mean + sigmoid contrastive loss (scalar out).
// ---------------------------------------------------------------------------
__global__ void k_loss(const float* __restrict__ maxbuf,
                       const int* __restrict__ lens,
                       const float* __restrict__ logit_scale,
                       const float* __restrict__ logit_bias,
                       float* __restrict__ out) {
  __shared__ float red[1024];
  const int tid = threadIdx.x;
  const int mm = tid >> 5;
  const int cc = tid & 31;

  const int len = lens[cc];
  const float* row = maxbuf + ((size_t)mm * B_ + cc) * W_;
  float s = 0.0f;
  for (int w = 0; w < len; ++w) s += row[w];
  const float sim = s / (float)len;

  const float t = __expf(fminf(fmaxf(logit_scale[0], -10.0f), 10.0f));
  const float logit = fminf(fmaxf(t * sim + logit_bias[0], -50.0f), 50.0f);
  const float z = (mm == cc) ? 1.0f : -1.0f;
  const float x = z * logit;
  const float lsig = fminf(x, 0.0f) - log1pf(__expf(-fabsf(x)));

  red[tid] = lsig;
  __syncthreads();
#pragma unroll
  for (int off = 512; off >= 1; off >>= 1) {
    if (tid < off) red[tid] += red[tid + off];
    __syncthreads();
  }
  if (tid == 0) out[0] = -red[0] / (float)(B_ * B_);
}

// ---------------------------------------------------------------------------
extern "C" void kernel_launch(void* const* d_in, const int* in_sizes, int n_in,
                              void* d_out, int out_size, void* d_ws, size_t ws_size,
                              hipStream_t stream) {
  const float* img  = (const float*)d_in[0];  // (B,N,D) f32
  const float* con  = (const float*)d_in[1];  // (B,W,D) f32
  const int*   lens = (const int*)d_in[2];    // (B,)    i32
  const float* ls   = (const float*)d_in[3];  // (1,)    f32
  const float* lb   = (const float*)d_in[4];  // (1,)    f32

  float* img_n  = (float*)d_ws;
  float* con_n  = img_n + (size_t)B_ * N_ * D_;
  float* maxbuf = con_n + (size_t)B_ * W_ * D_;

  const int totalRows = B_ * N_ + B_ * W_;  // 19456 rows
  const int rowsPerBlk = 8;                 // 256 threads = 8 waves
  const int blocks = (totalRows + rowsPerBlk - 1) / rowsPerBlk;
  k_normalize<<<blocks, 256, 0, stream>>>(img, con, img_n, con_n);

  k_gemm_max<<<B_ * B_, 128, 0, stream>>>(img_n, con_n, maxbuf);

  k_loss<<<1, 1024, 0, stream>>>(maxbuf, lens, ls, lb, (float*)d_out);
}